// RiemannianIFNode_25202868093199
// MI455X (gfx1250) — compile-verified
//
#include <hip/hip_runtime.h>
#include <hip/hip_bf16.h>
#include <cstddef>

typedef __attribute__((ext_vector_type(2))) float v2f;
typedef __attribute__((ext_vector_type(4))) float v4f;
typedef __attribute__((ext_vector_type(8))) float v8f;

#define T_STEPS 128
#define B_DIM   2048
#define D_DIM   256
#define E_TOT   (B_DIM * D_DIM)      // 524288 independent channels
#define CHUNK   16
#define NCHUNK  (T_STEPS / CHUNK)    // 8
#define WAVES_PER_BLOCK 2
#define ELEMS_PER_WAVE  32           // two 16-column WMMA N-tiles per wave
#define YSTRIDE 132                  // 128 + pad, keeps 16B alignment (132*4 % 16 == 0)
#define CSTRIDE 20                   // 16 + pad, keeps 16B alignment (20*4 % 16 == 0)

// ---------------------------------------------------------------------------
// GL fractional-order IF scan.
// Per wave: 32 element channels, all 128 time steps.
// Cross-chunk history term via V_WMMA_F32_16X16X4_F32 (fp32 WMMA, 4 K-slices
// per 16x16x16 product); intra-chunk 16-step triangle on VALU.
// History + transpose stage stored COLUMN-major in LDS so B fragments are
// ds_load_b64 and history commits / cross reads are ds_*_b128. A fragments
// read from a REVERSED coefficient table so the ascending 2addr LDS load
// fills the v2f pair in WMMA operand order (no register swaps).
// ---------------------------------------------------------------------------
__global__ __launch_bounds__(WAVES_PER_BLOCK * 32)
void glif_scan_kernel(const float* __restrict__ x, float* __restrict__ out)
{
    __shared__ float c_sh[132];                                    // c[0..128]
    __shared__ float crev[132];                                    // crev[j] = c[127-j]
    __shared__ float Yt[WAVES_PER_BLOCK][ELEMS_PER_WAVE][YSTRIDE]; // history, [col][t]
    __shared__ float CSt[WAVES_PER_BLOCK][ELEMS_PER_WAVE][CSTRIDE];// cross stage, [col][t]

    const int tid = threadIdx.x;
    if (tid == 0) {
        // c[0]=1; c[j]=(1-(1+alpha)/j)*c[j-1], alpha=0.5, fp64 like the reference
        double cd = 1.0;
        c_sh[0] = 1.0f;
        for (int j = 1; j <= T_STEPS; ++j) {
            cd *= (1.0 - 1.5 / (double)j);
            c_sh[j] = (float)cd;
        }
        for (int j = 0; j < T_STEPS; ++j)
            crev[j] = c_sh[127 - j];
    }
    __syncthreads();

    const int wv   = tid >> 5;
    const int lane = tid & 31;
    const int tt   = lane & 15;   // row (A) / column (B) within a 16-tile
    const int hi   = lane >> 4;   // half-wave select
    const size_t e = (size_t)blockIdx.x * (WAVES_PER_BLOCK * ELEMS_PER_WAVE)
                   + (size_t)wv * ELEMS_PER_WAVE + lane;

    // intra-chunk triangle coefficients c[1..15] in registers
    float cin[CHUNK];
#pragma unroll
    for (int j = 0; j < CHUNK; ++j) cin[j] = c_sh[j];

    for (int m = 0; m < NCHUNK; ++m) {
        // This chunk's inputs (streamed once: non-temporal)
        float xv[CHUNK];
#pragma unroll
        for (int t = 0; t < CHUNK; ++t)
            xv[t] = __builtin_nontemporal_load(&x[(size_t)(m * CHUNK + t) * E_TOT + e]);

        // Software-prefetch next chunk's x rows while the WMMAs run (uniform branch)
        if (m + 1 < NCHUNK) {
#pragma unroll
            for (int t = 0; t < CHUNK; t += 4)
                __builtin_prefetch(&x[(size_t)((m + 1) * CHUNK + t) * E_TOT + e], 0, 0);
        }

        // ---- cross-chunk history term via fp32 WMMA ----
        v8f acc0 = {};   // columns 0..15 of this wave's 32-element tile
        v8f acc1 = {};   // columns 16..31
        for (int d = 1; d <= m; ++d) {          // uniform loop: EXEC stays all-ones
            const int p = m - d;
#pragma unroll
            for (int q = 0; q < 4; ++q) {       // K = 16 split into 4 x K4 slices
                const int k0 = 4 * q + 2 * hi;  // even -> 8B-aligned b64 loads
                // A[t,s] = c[16d + t - s]; reversed table -> ascending 2addr load
                const int ridx = 127 - (16 * d + tt - k0);
                v2f a;
                a.x = crev[ridx];               // = c[16d + tt - k0]
                a.y = crev[ridx + 1];           // = c[16d + tt - k0 - 1]
                // B = Y chunk p, rows k0..k0+1 at this lane's column: one b64 each
                v2f b0 = *(const v2f*)&Yt[wv][tt][CHUNK * p + k0];
                v2f b1 = *(const v2f*)&Yt[wv][tt + 16][CHUNK * p + k0];
                acc0 = __builtin_amdgcn_wmma_f32_16x16x4_f32(
                    false, a, false, b0, (short)0, acc0, false, false);
                acc1 = __builtin_amdgcn_wmma_f32_16x16x4_f32(
                    false, a, false, b1, (short)0, acc1, false, false);
            }
        }

        // ---- spill D fragments [t x e] -> column-major stage (b128 stores) ----
        {
            const int rbase = 8 * hi;           // C/D layout: VGPR r = rows r / r+8
            v4f s00 = {acc0[0], acc0[1], acc0[2], acc0[3]};
            v4f s01 = {acc0[4], acc0[5], acc0[6], acc0[7]};
            v4f s10 = {acc1[0], acc1[1], acc1[2], acc1[3]};
            v4f s11 = {acc1[4], acc1[5], acc1[6], acc1[7]};
            *(v4f*)&CSt[wv][tt][rbase]          = s00;
            *(v4f*)&CSt[wv][tt][rbase + 4]      = s01;
            *(v4f*)&CSt[wv][tt + 16][rbase]     = s10;
            *(v4f*)&CSt[wv][tt + 16][rbase + 4] = s11;
        }
        __syncthreads();

        // ---- per-lane cross term readback (b128 loads) ----
        float crossv[CHUNK];
#pragma unroll
        for (int t4 = 0; t4 < CHUNK; t4 += 4)
            *(v4f*)&crossv[t4] = *(const v4f*)&CSt[wv][lane][t4];

        // ---- intra-chunk sequential IF scan (per-lane, fully unrolled) ----
        float ys[CHUNK];
#pragma unroll
        for (int t = 0; t < CHUNK; ++t) {
            float men = xv[t] - crossv[t];
#pragma unroll
            for (int s = 0; s < t; ++s)
                men -= cin[t - s] * ys[s];
            const float spike = (men > 1.0f) ? 1.0f : 0.0f;
            men -= spike;                       // VTH = 1
            ys[t] = men;
            __builtin_nontemporal_store(spike,
                &out[(size_t)(m * CHUNK + t) * E_TOT + e]);
        }

        // ---- commit this chunk's history, column-major (b128 stores) ----
#pragma unroll
        for (int t4 = 0; t4 < CHUNK; t4 += 4) {
            v4f yw = {ys[t4], ys[t4 + 1], ys[t4 + 2], ys[t4 + 3]};
            *(v4f*)&Yt[wv][lane][m * CHUNK + t4] = yw;
        }
        __syncthreads();                        // fences Yt reads + CSt reuse
    }
}

// ---------------------------------------------------------------------------
// Lorentz exponential map: one 256-thread block per hyperboloid row.
// ---------------------------------------------------------------------------
__global__ __launch_bounds__(256)
void lorentz_expmap_kernel(const float* __restrict__ v, const float* __restrict__ z,
                           float* __restrict__ out)
{
    __shared__ float red[256];
    const int b = blockIdx.x;
    const int i = threadIdx.x;
    const float vi = v[(size_t)b * D_DIM + i];
    float term = vi * vi;
    if (i == 0) term = -term;                   // Minkowski: -v0^2 + ||v_1:||^2
    red[i] = term;
    __syncthreads();
    for (int off = 128; off > 0; off >>= 1) {
        if (i < off) red[i] += red[i + off];
        __syncthreads();
    }
    const float vv = red[0];
    const float s  = sqrtf(fmaxf(vv, 1e-6f));
    const float zi = z[(size_t)b * D_DIM + i];
    __builtin_nontemporal_store(coshf(s) * zi + (sinhf(s) / s) * vi,
        &out[(size_t)T_STEPS * E_TOT + (size_t)b * D_DIM + i]);
}

extern "C" void kernel_launch(void* const* d_in, const int* in_sizes, int n_in,
                              void* d_out, int out_size, void* d_ws, size_t ws_size,
                              hipStream_t stream)
{
    (void)in_sizes; (void)n_in; (void)d_ws; (void)ws_size; (void)out_size;
    const float* x = (const float*)d_in[0];   // x_seq [T,B,D]
    const float* v = (const float*)d_in[1];   // v_seq [B,D]
    const float* z = (const float*)d_in[2];   // z_seq [B,D]
    float* out = (float*)d_out;               // s_seq [T,B,D] ++ z_out [B,D]

    const int blocks = E_TOT / (WAVES_PER_BLOCK * ELEMS_PER_WAVE); // 8192
    glif_scan_kernel<<<blocks, WAVES_PER_BLOCK * 32, 0, stream>>>(x, out);
    lorentz_expmap_kernel<<<B_DIM, D_DIM, 0, stream>>>(v, z, out);
}